// FingerNetWrapper_10496900071827
// MI455X (gfx1250) — compile-verified
//
#include <hip/hip_runtime.h>
#include <stdint.h>

// ---------------------------------------------------------------------------
// FingerNet wrapper for MI455X (gfx1250, wave32).
// Outputs (concat, float32): minut[4,1024,4] | keep[4,1024] | enh_vis[4,512,512]
//                            | mask_up*255[4,512,512] | ori_field[4,512,512]
// Workspace use: ~9.2 MB.
// CDNA5 paths: v_wmma_f32_16x16x4_f32 (pairwise NMS distances),
//              global_load_async_to_lds_b128 + s_wait_asynccnt (detect staging),
//              global_prefetch (orientation streaming).
// ---------------------------------------------------------------------------

#define BB     4
#define HS     64
#define WSZ    64
#define HH     512
#define WW     512
#define C_ORI  90
#define C_MNT  180
#define C_OFF  8
#define KK     1024
#define NCELL  (HS * WSZ)     // 4096
#define HW     (HH * WW)      // 262144
#define PI_F   3.14159265358979323846f
#define ANG_T  (PI_F / 6.0f)

typedef __attribute__((ext_vector_type(2))) float v2f;
typedef __attribute__((ext_vector_type(8))) float v8f;
typedef __attribute__((ext_vector_type(4))) int   v4i;

#define AS_GLOBAL __attribute__((address_space(1)))
#define AS_LDS    __attribute__((address_space(3)))

// ---- workspace layout (float offsets) -------------------------------------
#define OFF_MASK_S  ((size_t)0)             // 16384
#define OFF_TMP_S   ((size_t)16384)         // 16384
#define OFF_MASK_UP ((size_t)32768)         // 1048576
#define OFF_TMP_UP  ((size_t)1081344)       // 1048576
#define OFF_CANDX   ((size_t)2129920)       // 4096
#define OFF_CANDY   ((size_t)2134016)
#define OFF_CANDA   ((size_t)2138112)
#define OFF_CANDS   ((size_t)2142208)
#define OFF_CANDV   ((size_t)2146304)
#define OFF_KEEPF   ((size_t)2150400)       // 4096
#define OFF_MM      ((size_t)2154496)       // 8
#define OFF_SUP     ((size_t)2154512)       // uint32[4*1024*32] = 131072 words

// ---- output layout (float offsets) ----------------------------------------
#define OUT0 ((size_t)0)        // minut  16384
#define OUT1 ((size_t)16384)    // keep    4096
#define OUT2 ((size_t)20480)    // enh_vis 1048576
#define OUT3 ((size_t)1069056)  // mask_up*255
#define OUT4 ((size_t)2117632)  // ori_field

// ---------------------------------------------------------------------------
// Separable morphological pass: min (erode) or max (dilate) over a 1-D window
// [-pl, +pr] along x (axis=0) or y (axis=1). Out-of-range = neutral (matches
// reference padding with +/-1e4). Optionally rounds input (half-to-even).
// ---------------------------------------------------------------------------
__global__ void morph_pass_kernel(const float* __restrict__ in, float* __restrict__ out,
                                  int Wd, int Hd, int nimg, int axis, int pl, int pr,
                                  int isMax, int doRound) {
    int idx = blockIdx.x * blockDim.x + threadIdx.x;
    int total = nimg * Hd * Wd;
    if (idx >= total) return;
    int x = idx % Wd;
    int y = (idx / Wd) % Hd;
    int b = idx / (Wd * Hd);
    const float* img = in + (size_t)b * Hd * Wd;
    float best = isMax ? -1e30f : 1e30f;
    for (int d = -pl; d <= pr; ++d) {
        int xx = x, yy = y;
        if (axis == 0) xx = x + d; else yy = y + d;
        if (xx < 0 || xx >= Wd || yy < 0 || yy >= Hd) continue;
        float v = img[yy * Wd + xx];
        if (doRound) v = rintf(v);
        best = isMax ? fmaxf(best, v) : fminf(best, v);
    }
    out[idx] = best;
}

// ---------------------------------------------------------------------------
// Bandwidth-dominant kernel: argmax over 90 channels (377 MB stream), fused
// with writing mask_up*255 and ori_field. float4 per thread for b128 loads,
// prefetch of the next channel plane.
// ---------------------------------------------------------------------------
__global__ void ori_mask_kernel(const float* __restrict__ ori, const float* __restrict__ maskUp,
                                float* __restrict__ out3, float* __restrict__ out4) {
    int t = blockIdx.x * blockDim.x + threadIdx.x;
    int total = BB * HW / 4;
    if (t >= total) return;
    int p = t * 4;
    int b = p / HW;
    int pix = p - b * HW;
    const float* base = ori + (size_t)b * C_ORI * HW + pix;

    float4 v0 = *(const float4*)base;
    float b0 = v0.x, b1 = v0.y, b2 = v0.z, b3 = v0.w;
    int i0 = 0, i1 = 0, i2 = 0, i3 = 0;
    for (int c = 1; c < C_ORI; ++c) {
        const float4 v = *(const float4*)(base + (size_t)c * HW);
        if (c + 1 < C_ORI) __builtin_prefetch(base + (size_t)(c + 1) * HW, 0, 1);
        if (v.x > b0) { b0 = v.x; i0 = c; }
        if (v.y > b1) { b1 = v.y; i1 = c; }
        if (v.z > b2) { b2 = v.z; i2 = c; }
        if (v.w > b3) { b3 = v.w; i3 = c; }
    }
    float4 m = *(const float4*)(maskUp + p);
    const float s = PI_F / 180.0f;
    float4 o;
    o.x = ((float)i0 * 2.0f - 90.0f) * s * m.x;
    o.y = ((float)i1 * 2.0f - 90.0f) * s * m.y;
    o.z = ((float)i2 * 2.0f - 90.0f) * s * m.z;
    o.w = ((float)i3 * 2.0f - 90.0f) * s * m.w;
    *(float4*)(out4 + p) = o;
    float4 mo = make_float4(m.x * 255.0f, m.y * 255.0f, m.z * 255.0f, m.w * 255.0f);
    *(float4*)(out3 + p) = mo;
}

// ---------------------------------------------------------------------------
// Per-image min/max of enhanced_real * mask_up.
// ---------------------------------------------------------------------------
__global__ void enh_reduce_kernel(const float* __restrict__ enh, const float* __restrict__ maskUp,
                                  float* __restrict__ mm) {
    __shared__ float smin[256];
    __shared__ float smax[256];
    int b = blockIdx.x;
    int t = threadIdx.x;
    const float* e = enh + (size_t)b * HW;
    const float* m = maskUp + (size_t)b * HW;
    float mn = 1e30f, mx = -1e30f;
    for (int p = t; p < HW; p += 256) {
        float v = e[p] * m[p];
        mn = fminf(mn, v);
        mx = fmaxf(mx, v);
    }
    smin[t] = mn; smax[t] = mx;
    __syncthreads();
    for (int s = 128; s > 0; s >>= 1) {
        if (t < s) {
            smin[t] = fminf(smin[t], smin[t + s]);
            smax[t] = fmaxf(smax[t], smax[t + s]);
        }
        __syncthreads();
    }
    if (t == 0) { mm[b * 2] = smin[0]; mm[b * 2 + 1] = smax[0]; }
}

__global__ void enh_norm_kernel(const float* __restrict__ enh, const float* __restrict__ maskUp,
                                const float* __restrict__ mm, float* __restrict__ out2) {
    int t = blockIdx.x * blockDim.x + threadIdx.x;
    if (t >= BB * HW) return;
    int b = t / HW;
    float emin = mm[b * 2], emax = mm[b * 2 + 1];
    float v = enh[t] * maskUp[t];
    out2[t] = (v - emin) / (emax - emin + 1e-8f) * 255.0f;
}

// ---------------------------------------------------------------------------
// Per-image detection: score/mask planes staged to LDS with the gfx1250 async
// engine (GLOBAL_LOAD_ASYNC_TO_LDS_B128, ASYNCcnt), then bitonic sort of 4096
// (score desc, index asc) keys in LDS reproduces jax.lax.top_k ordering +
// tie-breaking; then per-candidate channel argmaxes (180 orientation, 8+8
// offsets). The staging buffers alias the key buffer (read-out before keys
// are written, separated by a barrier).
// ---------------------------------------------------------------------------
__global__ void detect_kernel(const float* __restrict__ score, const float* __restrict__ maskS,
                              const float* __restrict__ mori, const float* __restrict__ mxo,
                              const float* __restrict__ myo,
                              float* __restrict__ candX, float* __restrict__ candY,
                              float* __restrict__ candA, float* __restrict__ candS,
                              float* __restrict__ candV) {
    __shared__ unsigned long long keys[NCELL];   // 32 KB (first 32 KB also used as staging)
    int b = blockIdx.x;
    int t = threadIdx.x;
    const float* sc = score + (size_t)b * NCELL;
    const float* mk = maskS + (size_t)b * NCELL;

    // --- async stage: score and mask planes -> LDS (16 B per lane, aligned)
    float* sScore = (float*)keys;          // 4096 floats
    float* sMask  = ((float*)keys) + NCELL;
    {
        int e0 = t * 4;
        __builtin_amdgcn_global_load_async_to_lds_b128(
            (AS_GLOBAL v4i*)(sc + e0), (AS_LDS v4i*)(sScore + e0), 0, 0);
        __builtin_amdgcn_global_load_async_to_lds_b128(
            (AS_GLOBAL v4i*)(mk + e0), (AS_LDS v4i*)(sMask + e0), 0, 0);
        __builtin_amdgcn_s_wait_asynccnt(0);
    }
    __syncthreads();

    float sv[4];
#pragma unroll
    for (int q = 0; q < 4; ++q) {
        int e = t + q * 1024;
        sv[q] = sScore[e] * sMask[e];
    }
    __syncthreads();   // all reads of staging done before keys overwrite it

#pragma unroll
    for (int q = 0; q < 4; ++q) {
        int e = t + q * 1024;
        float masked = (sv[q] > 0.5f) ? sv[q] : -1.0f;
        unsigned u = __float_as_uint(masked);
        unsigned srt = (u & 0x80000000u) ? ~u : (u | 0x80000000u);  // ascending map
        unsigned ds = ~srt;                                          // descending
        keys[e] = ((unsigned long long)ds << 32) | (unsigned)e;
    }
    __syncthreads();

    for (unsigned k = 2; k <= NCELL; k <<= 1) {
        for (unsigned jj = k >> 1; jj > 0; jj >>= 1) {
            for (unsigned e = (unsigned)t; e < NCELL; e += 1024) {
                unsigned ixj = e ^ jj;
                if (ixj > e) {
                    bool up = ((e & k) == 0);
                    unsigned long long a = keys[e], c2 = keys[ixj];
                    if ((a > c2) == up) { keys[e] = c2; keys[ixj] = a; }
                }
            }
            __syncthreads();
        }
    }

    // rank t candidate
    unsigned long long kk = keys[t];
    int cell = (int)(unsigned)(kk & 0xFFFFFFFFull);
    float s = sc[cell] * mk[cell];
    int row = cell / WSZ, col = cell % WSZ;

    const float* ob = mori + (size_t)b * C_MNT * NCELL + cell;
    float bv = ob[0]; int bi = 0;
    for (int c = 1; c < C_MNT; ++c) {
        float v = ob[(size_t)c * NCELL];
        if (v > bv) { bv = v; bi = c; }
    }
    const float* xb = mxo + (size_t)b * C_OFF * NCELL + cell;
    float xv = xb[0]; int xi = 0;
    for (int c = 1; c < C_OFF; ++c) { float v = xb[(size_t)c * NCELL]; if (v > xv) { xv = v; xi = c; } }
    const float* yb = myo + (size_t)b * C_OFF * NCELL + cell;
    float yv = yb[0]; int yi = 0;
    for (int c = 1; c < C_OFF; ++c) { float v = yb[(size_t)c * NCELL]; if (v > yv) { yv = v; yi = c; } }

    int o = b * KK + t;
    candX[o] = (float)col * 8.0f + (float)xi;
    candY[o] = (float)row * 8.0f + (float)yi;
    candA[o] = ((float)bi * 2.0f - 89.0f) * (PI_F / 180.0f);
    candS[o] = s;
    candV[o] = (s > 0.5f) ? 1.0f : 0.0f;
}

// ---------------------------------------------------------------------------
// Pairwise suppression matrix via V_WMMA_F32_16X16X4_F32.
// d2_ij = a_i . b_j with a_i=(x_i, y_i, |p_i|^2, 1), b_j=(-2x_j, -2y_j, 1, |p_j|^2).
// One wave per 16x16 tile; branchless epilogue (hoisted loads + bitwise &),
// packs per-column 16-bit masks into supT (uint32[img][j][32]: word c bit t
// -> suppressor i = 32c+t).
// ---------------------------------------------------------------------------
__global__ void sup_wmma_kernel(const float* __restrict__ candX, const float* __restrict__ candY,
                                const float* __restrict__ candA, const float* __restrict__ candV,
                                uint32_t* __restrict__ supT) {
    __shared__ unsigned char tile[16][16];  // tile[j_local][i_local]
    int b = blockIdx.z;
    int i0 = blockIdx.y * 16;
    int j0 = blockIdx.x * 16;
    int lane = threadIdx.x;
    const float* X = candX + b * KK;
    const float* Y = candY + b * KK;
    const float* A = candA + b * KK;
    const float* V = candV + b * KK;

    // A 16x4 layout: lanes 0-15 hold K=0,1 for M=lane; lanes 16-31 hold K=2,3.
    // B 4x16 layout: lanes 0-15 hold K=0,1 for N=lane; lanes 16-31 hold K=2,3.
    int lo = lane & 15;
    int hi = (lane >> 4) & 1;           // 0: K=0,1  1: K=2,3
    int ia = i0 + lo;
    int ja = j0 + lo;
    float xi = X[ia], yi = Y[ia];
    float xj = X[ja], yj = Y[ja];
    v2f av, bvv;
    av.x  = hi ? (xi * xi + yi * yi) : xi;            // K0: x_i   | K2: |p_i|^2
    av.y  = hi ? 1.0f                : yi;            // K1: y_i   | K3: 1
    bvv.x = hi ? 1.0f                : (-2.0f * xj);  // K0: -2x_j | K2: 1
    bvv.y = hi ? (xj * xj + yj * yj) : (-2.0f * yj);  // K1: -2y_j | K3: |p_j|^2

    v8f cz = {0.0f, 0.0f, 0.0f, 0.0f, 0.0f, 0.0f, 0.0f, 0.0f};
    v8f d2 = __builtin_amdgcn_wmma_f32_16x16x4_f32(false, av, false, bvv,
                                                   (short)0, cz, false, false);

    // D layout: VGPR r holds (M = r + (lane<16?0:8), N = lane&15)
    int j = j0 + lo;
    float aj = A[j];
    float vj = V[j];
    int mbase = hi ? 8 : 0;

    float ai[8], vi[8];
#pragma unroll
    for (int r = 0; r < 8; ++r) {
        int i = i0 + mbase + r;
        ai[r] = A[i];
        vi[r] = V[i];
    }
#pragma unroll
    for (int r = 0; r < 8; ++r) {
        float da = fabsf(ai[r] - aj);
        da = fminf(da, 2.0f * PI_F - da);
        int pass = (int)(d2[r] < 256.0f) & (int)(da < ANG_T) & (int)(vi[r] * vj > 0.5f);
        tile[lo][mbase + r] = (unsigned char)pass;
    }
    __syncthreads();
    if (lane < 16) {
        unsigned bits = 0;
#pragma unroll
        for (int il = 0; il < 16; ++il) bits |= ((unsigned)tile[lane][il]) << il;
        unsigned short* sp = (unsigned short*)supT;
        sp[((size_t)b * KK + (size_t)(j0 + lane)) * 64 + (size_t)(i0 >> 4)] = (unsigned short)bits;
    }
}

// ---------------------------------------------------------------------------
// Sequential NMS, chunked: per 32-candidate chunk, the owning wave resolves
// the intra-chunk order with __shfl; then all later threads apply the whole
// chunk at once via a ballot-built keep mask. 64 barriers total.
// ---------------------------------------------------------------------------
__global__ void nms_kernel(const uint32_t* __restrict__ supT, const float* __restrict__ candV,
                           float* __restrict__ keepF) {
    __shared__ unsigned keepLDS[KK];
    int b = blockIdx.x;
    unsigned j = threadIdx.x;
    unsigned kj = (candV[b * KK + j] > 0.5f) ? 1u : 0u;
    keepLDS[j] = kj;
    __syncthreads();
    const uint32_t* srow = supT + ((size_t)b * KK + j) * 32;
    for (int c = 0; c < 32; ++c) {
        unsigned base = (unsigned)c * 32u;
        if ((j >> 5) == (unsigned)c) {
            unsigned w = srow[c];
            unsigned lane = j & 31u;
            int k = (int)kj;
            for (int t = 0; t < 32; ++t) {
                int ki = __shfl(k, t, 32);
                if (lane > (unsigned)t && ki && ((w >> (unsigned)t) & 1u)) k = 0;
            }
            kj = (unsigned)k;
            keepLDS[j] = kj;
        }
        __syncthreads();
        if (j >= base + 32u) {
            unsigned lane = j & 31u;
            unsigned keepmask = (unsigned)__ballot(keepLDS[base + lane] != 0u);
            unsigned w = srow[c];
            if (w & keepmask) kj = 0u;
        }
        __syncthreads();
    }
    keepF[b * KK + j] = kj ? 1.0f : 0.0f;
}

__global__ void finalize_kernel(const float* __restrict__ candX, const float* __restrict__ candY,
                                const float* __restrict__ candA, const float* __restrict__ candS,
                                const float* __restrict__ keepF,
                                float* __restrict__ out0, float* __restrict__ out1) {
    int t = blockIdx.x * blockDim.x + threadIdx.x;
    if (t >= BB * KK) return;
    float kf = keepF[t];
    out0[(size_t)t * 4 + 0] = candX[t] * kf;
    out0[(size_t)t * 4 + 1] = candY[t] * kf;
    out0[(size_t)t * 4 + 2] = candA[t] * kf;
    out0[(size_t)t * 4 + 3] = candS[t] * kf;
    out1[t] = kf;
}

// ---------------------------------------------------------------------------
extern "C" void kernel_launch(void* const* d_in, const int* in_sizes, int n_in,
                              void* d_out, int out_size, void* d_ws, size_t ws_size,
                              hipStream_t stream) {
    (void)in_sizes; (void)n_in; (void)out_size; (void)ws_size;
    const float* seg    = (const float*)d_in[0];  // [4,1,64,64]
    const float* segUp  = (const float*)d_in[1];  // [4,1,512,512]
    const float* oriUp  = (const float*)d_in[2];  // [4,90,512,512]
    const float* enh    = (const float*)d_in[3];  // [4,1,512,512]
    const float* mscore = (const float*)d_in[4];  // [4,1,64,64]
    const float* mori   = (const float*)d_in[5];  // [4,180,64,64]
    const float* mxo    = (const float*)d_in[6];  // [4,8,64,64]
    const float* myo    = (const float*)d_in[7];  // [4,8,64,64]

    float* w = (float*)d_ws;
    float* maskS  = w + OFF_MASK_S;
    float* tmpS   = w + OFF_TMP_S;
    float* maskUp = w + OFF_MASK_UP;
    float* tmpUp  = w + OFF_TMP_UP;
    float* candX  = w + OFF_CANDX;
    float* candY  = w + OFF_CANDY;
    float* candA  = w + OFF_CANDA;
    float* candS  = w + OFF_CANDS;
    float* candV  = w + OFF_CANDV;
    float* keepF  = w + OFF_KEEPF;
    float* mm     = w + OFF_MM;
    uint32_t* supT = (uint32_t*)(w + OFF_SUP);

    float* out = (float*)d_out;
    float* out0 = out + OUT0;
    float* out1 = out + OUT1;
    float* out2 = out + OUT2;
    float* out3 = out + OUT3;
    float* out4 = out + OUT4;

    // --- small mask: opening(round(seg), 5) : erode x, erode y, dilate x, dilate y
    {
        int N = BB * HS * WSZ, g = (N + 255) / 256;
        morph_pass_kernel<<<g, 256, 0, stream>>>(seg,   tmpS,  WSZ, HS, BB, 0, 2, 2, 0, 1);
        morph_pass_kernel<<<g, 256, 0, stream>>>(tmpS,  maskS, WSZ, HS, BB, 1, 2, 2, 0, 0);
        morph_pass_kernel<<<g, 256, 0, stream>>>(maskS, tmpS,  WSZ, HS, BB, 0, 2, 2, 1, 0);
        morph_pass_kernel<<<g, 256, 0, stream>>>(tmpS,  maskS, WSZ, HS, BB, 1, 2, 2, 1, 0);
    }
    // --- big mask: opening(round(seg_up), 40): pl=19, pr=20
    {
        int N = BB * HW, g = (N + 255) / 256;
        morph_pass_kernel<<<g, 256, 0, stream>>>(segUp,  tmpUp,  WW, HH, BB, 0, 19, 20, 0, 1);
        morph_pass_kernel<<<g, 256, 0, stream>>>(tmpUp,  maskUp, WW, HH, BB, 1, 19, 20, 0, 0);
        morph_pass_kernel<<<g, 256, 0, stream>>>(maskUp, tmpUp,  WW, HH, BB, 0, 19, 20, 1, 0);
        morph_pass_kernel<<<g, 256, 0, stream>>>(tmpUp,  maskUp, WW, HH, BB, 1, 19, 20, 1, 0);
    }
    // --- orientation argmax + mask_up outputs (bandwidth-dominant)
    {
        int N = BB * HW / 4, g = (N + 255) / 256;
        ori_mask_kernel<<<g, 256, 0, stream>>>(oriUp, maskUp, out3, out4);
    }
    // --- enhance normalize
    enh_reduce_kernel<<<BB, 256, 0, stream>>>(enh, maskUp, mm);
    {
        int N = BB * HW, g = (N + 255) / 256;
        enh_norm_kernel<<<g, 256, 0, stream>>>(enh, maskUp, mm, out2);
    }
    // --- minutiae detection (async LDS staging + bitonic top-k + argmaxes)
    detect_kernel<<<BB, 1024, 0, stream>>>(mscore, maskS, mori, mxo, myo,
                                           candX, candY, candA, candS, candV);
    // --- pairwise suppression matrix via WMMA f32 16x16x4
    {
        dim3 grid(KK / 16, KK / 16, BB);
        sup_wmma_kernel<<<grid, 32, 0, stream>>>(candX, candY, candA, candV, supT);
    }
    // --- sequential NMS (chunked scan)
    nms_kernel<<<BB, 1024, 0, stream>>>(supT, candV, keepF);
    // --- final outputs
    {
        int N = BB * KK, g = (N + 255) / 256;
        finalize_kernel<<<g, 256, 0, stream>>>(candX, candY, candA, candS, keepF, out0, out1);
    }
}